// MaskedMHCA_72765335929374
// MI455X (gfx1250) — compile-verified
//
#include <hip/hip_runtime.h>
#include <hip/hip_bf16.h>
#include <math.h>

// ---------------------------------------------------------------------------
// MaskedMHCA for MI455X (gfx1250, wave32).
//   EMBD_DIM C=1024, N_HEADS H=16, head dim D=64, WINDOW W=9, STRIDE S=4,
//   BS=4, T=4096.  Projections use V_WMMA_F32_16X16X4_F32 (true fp32 WMMA)
//   with 2x2 register blocking (32x32 output tile per wave) so the K-loop
//   issues 4 WMMAs per 4-6 VMEM loads instead of 1 per 3.
// ---------------------------------------------------------------------------

#define C_DIM   1024
#define T_DIM   4096
#define BS_DIM  4
#define H_DIM   16
#define D_DIM   64
#define NT      (BS_DIM * T_DIM)          // 16384 token positions
#define SCALE   0.125f                    // 1/sqrt(64)

typedef float v2f __attribute__((ext_vector_type(2)));
typedef float v8f __attribute__((ext_vector_type(8)));

// ---------------------------------------------------------------------------
// Kernel 1: fused QKV projection.
//   y[o, bt] = sum_c W[o,c] * x[c, bt] + b[o]     (x = q in (BS, C, T) layout)
// Output in head-major layout: dst[(b*H + o/64)][t][o%64], q scaled by 1/sqrt(d).
// One wave computes a 32(M) x 32(N) tile as 2x2 WMMA subtiles; 8 waves per
// block share (proj, mtile) with consecutive ntiles so A-loads hit L0/L1.
// ---------------------------------------------------------------------------
__global__ __launch_bounds__(256) void mhca_qkv_proj(
    const float* __restrict__ q,
    const float* __restrict__ Wq, const float* __restrict__ bq,
    const float* __restrict__ Wk, const float* __restrict__ bk,
    const float* __restrict__ Wv, const float* __restrict__ bv,
    float* __restrict__ QH, float* __restrict__ KH, float* __restrict__ VH)
{
    const int lane = threadIdx.x & 31;
    const int wv   = threadIdx.x >> 5;
    const unsigned tileId = blockIdx.x * 8u + wv;     // 3 * 32 * 512 tiles
    const int ntile = tileId & 511;                   // 512 N tiles (32 tokens each)
    const int rest  = tileId >> 9;
    const int mtile = rest & 31;                      // 32 M tiles (32 channels each)
    const int proj  = rest >> 5;                      // 0=q, 1=k, 2=v

    const float* Wm = (proj == 0) ? Wq : (proj == 1) ? Wk : Wv;
    const float* bm = (proj == 0) ? bq : (proj == 1) ? bk : bv;
    float*       dst = (proj == 0) ? QH : (proj == 1) ? KH : VH;

    const int bt0 = ntile * 32;
    const int b   = bt0 >> 12;                        // bt0 / T
    const int t0  = bt0 & (T_DIM - 1);
    const int m0  = mtile * 32;
    const int lm  = lane & 15;                        // A: M row lane / B: N col lane
    const int kl  = (lane >> 4) * 2;                  // K sub-offset per half-wave

    // A: W[m0(+16)+lm, k+kl .. k+kl+1]  (contiguous pair -> b64 load)
    const float* A0row = Wm + (size_t)(m0 + lm) * C_DIM + kl;
    const float* A1row = A0row + (size_t)16 * C_DIM;
    // B: x[k+kl, bt0(+16)+lm] = q[((b*C + k+kl) * T) + t0(+16)+lm]
    const float* Bcol = q + ((size_t)(b * C_DIM + kl)) * T_DIM + (t0 + lm);

    v8f acc00 = {}, acc01 = {}, acc10 = {}, acc11 = {};
#pragma unroll 4
    for (int k = 0; k < C_DIM; k += 4) {
        const float2 a0v = *(const float2*)(A0row + k);
        const float2 a1v = *(const float2*)(A1row + k);
        v2f a0; a0[0] = a0v.x; a0[1] = a0v.y;
        v2f a1; a1[0] = a1v.x; a1[1] = a1v.y;
        const float* bp0 = Bcol + (size_t)k * T_DIM;
        const float* bp1 = bp0 + (size_t)T_DIM;
        v2f b0; b0[0] = bp0[0];  b0[1] = bp1[0];      // tokens t0+lm
        v2f b1; b1[0] = bp0[16]; b1[1] = bp1[16];     // tokens t0+16+lm
        acc00 = __builtin_amdgcn_wmma_f32_16x16x4_f32(false, a0, false, b0, (short)0, acc00, false, false);
        acc01 = __builtin_amdgcn_wmma_f32_16x16x4_f32(false, a0, false, b1, (short)0, acc01, false, false);
        acc10 = __builtin_amdgcn_wmma_f32_16x16x4_f32(false, a1, false, b0, (short)0, acc10, false, false);
        acc11 = __builtin_amdgcn_wmma_f32_16x16x4_f32(false, a1, false, b1, (short)0, acc11, false, false);
    }

    const float scl = (proj == 0) ? SCALE : 1.0f;
    const int   mhi = (lane >> 4) << 3;               // +8 rows for upper half-wave
#pragma unroll
    for (int mi = 0; mi < 2; ++mi) {
        const v8f* am[2] = { &acc00, &acc10 };
        const v8f* an[2] = { &acc01, &acc11 };
#pragma unroll
        for (int r = 0; r < 8; ++r) {
            const int o    = m0 + mi * 16 + r + mhi;  // output channel
            const int head = o >> 6;
            const int dd   = o & 63;
            const float bias = bm[o];
            const size_t rowbase =
                ((size_t)(b * H_DIM + head)) * T_DIM * D_DIM + dd;
            const float v0 = ((*am[mi])[r] + bias) * scl;
            const float v1 = ((*an[mi])[r] + bias) * scl;
            dst[rowbase + (size_t)(t0 + lm) * D_DIM]      = v0;
            dst[rowbase + (size_t)(t0 + 16 + lm) * D_DIM] = v1;
        }
    }
}

// ---------------------------------------------------------------------------
// Kernel 2: windowed attention (w=9, offsets -4..+4), one wave per (n, t).
// Each lane owns 2 of the 64 head dims; scores via shfl_xor tree reduction.
// Output overwrites QH in place (q[n,t] is only consumed by this wave).
// ---------------------------------------------------------------------------
__global__ __launch_bounds__(256) void mhca_win_attn(
    float* __restrict__ QH, const float* __restrict__ KH,
    const float* __restrict__ VH, const unsigned char* __restrict__ qmask)
{
    const int lane = threadIdx.x & 31;
    const int wv   = threadIdx.x >> 5;
    const unsigned id = blockIdx.x * 8u + wv;         // 64 * 4096 waves
    const int n = id >> 12;                           // bs*h index
    const int t = id & (T_DIM - 1);
    const int b = n >> 4;

    const size_t nbase = (size_t)n * T_DIM * D_DIM;
    const int    dlo   = lane * 2;

    const float2 qv = *(const float2*)(QH + nbase + (size_t)t * D_DIM + dlo);

    float sc[9];
    int   ttc[9];
#pragma unroll
    for (int wi = 0; wi < 9; ++wi) {
        const int tt  = t + wi - 4;
        const bool ok = (tt >= 0) && (tt < T_DIM);
        const int tc  = ok ? tt : (tt < 0 ? 0 : T_DIM - 1);
        ttc[wi] = tc;
        const float2 kv = *(const float2*)(KH + nbase + (size_t)tc * D_DIM + dlo);
        float p = qv.x * kv.x + qv.y * kv.y;
#pragma unroll
        for (int off = 16; off >= 1; off >>= 1)
            p += __shfl_xor(p, off, 32);              // full 64-dim dot in all lanes
        const float bias = qmask[b * T_DIM + tc] ? 0.0f : -10000.0f;
        sc[wi] = ok ? (p + bias) : -INFINITY;
    }

    float m = sc[0];
#pragma unroll
    for (int wi = 1; wi < 9; ++wi) m = fmaxf(m, sc[wi]);
    float e[9];
    float s = 0.0f;
#pragma unroll
    for (int wi = 0; wi < 9; ++wi) { e[wi] = __expf(sc[wi] - m); s += e[wi]; }
    const float inv = 1.0f / s;
    const float qvalid = qmask[b * T_DIM + t] ? 1.0f : 0.0f;

    float2 o; o.x = 0.0f; o.y = 0.0f;
#pragma unroll
    for (int wi = 0; wi < 9; ++wi) {
        const float a = e[wi] * inv * qvalid;
        const float2 vv = *(const float2*)(VH + nbase + (size_t)ttc[wi] * D_DIM + dlo);
        o.x += a * vv.x;
        o.y += a * vv.y;
    }
    *(float2*)(QH + nbase + (size_t)t * D_DIM + dlo) = o;   // in-place
}

// ---------------------------------------------------------------------------
// Kernel 3: output projection.  out[b, o, t] = (Wp @ ao)[o, bt] + bp[o], masked.
// ao lives in head-major layout (the attention output, in QH's buffer).
// Same 2x2 WMMA tiling; here both A and B are b64 loads (4 loads / 4 WMMA).
// ---------------------------------------------------------------------------
__global__ __launch_bounds__(256) void mhca_out_proj(
    const float* __restrict__ AO, const float* __restrict__ Wp,
    const float* __restrict__ bp, const unsigned char* __restrict__ qmask,
    float* __restrict__ out)
{
    const int lane = threadIdx.x & 31;
    const int wv   = threadIdx.x >> 5;
    const unsigned tileId = blockIdx.x * 8u + wv;     // 32 * 512 tiles
    const int ntile = tileId & 511;
    const int mtile = tileId >> 9;

    const int bt0 = ntile * 32;
    const int b   = bt0 >> 12;
    const int t0  = bt0 & (T_DIM - 1);
    const int m0  = mtile * 32;
    const int lm  = lane & 15;
    const int kl  = (lane >> 4) * 2;

    const float* A0row = Wp + (size_t)(m0 + lm) * C_DIM + kl;
    const float* A1row = A0row + (size_t)16 * C_DIM;

    v8f acc00 = {}, acc01 = {}, acc10 = {}, acc11 = {};
#pragma unroll 4
    for (int k = 0; k < C_DIM; k += 4) {
        const float2 a0v = *(const float2*)(A0row + k);
        const float2 a1v = *(const float2*)(A1row + k);
        v2f a0; a0[0] = a0v.x; a0[1] = a0v.y;
        v2f a1; a1[0] = a1v.x; a1[1] = a1v.y;
        // B: x2[c, bt] = AO[(b*H + c/64)][t][c%64]; c = k+kl is even, so the
        // (c, c+1) pair is contiguous inside one head's 64-float chunk.
        const int c = k + kl;
        const float* bbase =
            AO + ((size_t)(b * H_DIM + (c >> 6))) * T_DIM * D_DIM + (c & 63);
        const float2 bv0 = *(const float2*)(bbase + (size_t)(t0 + lm) * D_DIM);
        const float2 bv1 = *(const float2*)(bbase + (size_t)(t0 + 16 + lm) * D_DIM);
        v2f b0; b0[0] = bv0.x; b0[1] = bv0.y;
        v2f b1; b1[0] = bv1.x; b1[1] = bv1.y;
        acc00 = __builtin_amdgcn_wmma_f32_16x16x4_f32(false, a0, false, b0, (short)0, acc00, false, false);
        acc01 = __builtin_amdgcn_wmma_f32_16x16x4_f32(false, a0, false, b1, (short)0, acc01, false, false);
        acc10 = __builtin_amdgcn_wmma_f32_16x16x4_f32(false, a1, false, b0, (short)0, acc10, false, false);
        acc11 = __builtin_amdgcn_wmma_f32_16x16x4_f32(false, a1, false, b1, (short)0, acc11, false, false);
    }

    const float mv0 = qmask[b * T_DIM + t0 + lm]      ? 1.0f : 0.0f;
    const float mv1 = qmask[b * T_DIM + t0 + 16 + lm] ? 1.0f : 0.0f;
    const int   mhi = (lane >> 4) << 3;
#pragma unroll
    for (int mi = 0; mi < 2; ++mi) {
        const v8f* am[2] = { &acc00, &acc10 };
        const v8f* an[2] = { &acc01, &acc11 };
#pragma unroll
        for (int r = 0; r < 8; ++r) {
            const int o = m0 + mi * 16 + r + mhi;
            const float bias = bp[o];
            const size_t rowbase = ((size_t)b * C_DIM + o) * T_DIM;
            out[rowbase + t0 + lm]      = ((*am[mi])[r] + bias) * mv0;
            out[rowbase + t0 + 16 + lm] = ((*an[mi])[r] + bias) * mv1;
        }
    }
}

// ---------------------------------------------------------------------------
// Kernel 4: second tuple output — q_mask as 0/1 in the output dtype (float).
// ---------------------------------------------------------------------------
__global__ void mhca_mask_copy(const unsigned char* __restrict__ qmask,
                               float* __restrict__ out_tail, int n)
{
    const int i = blockIdx.x * 256 + threadIdx.x;
    if (i < n) out_tail[i] = qmask[i] ? 1.0f : 0.0f;
}

// ---------------------------------------------------------------------------
extern "C" void kernel_launch(void* const* d_in, const int* in_sizes, int n_in,
                              void* d_out, int out_size, void* d_ws, size_t ws_size,
                              hipStream_t stream)
{
    const float*         q     = (const float*)d_in[0];
    const unsigned char* qmask = (const unsigned char*)d_in[1];
    const float* Wq = (const float*)d_in[2];
    const float* bq = (const float*)d_in[3];
    const float* Wk = (const float*)d_in[4];
    const float* bk = (const float*)d_in[5];
    const float* Wv = (const float*)d_in[6];
    const float* bv = (const float*)d_in[7];
    const float* Wp = (const float*)d_in[8];
    const float* bp = (const float*)d_in[9];
    float* out = (float*)d_out;

    const size_t plane = (size_t)NT * C_DIM;          // 16384 * 1024 floats (64 MB)
    float* QH = (float*)d_ws;                         // becomes attn output in place
    float* KH = QH + plane;
    float* VH = KH + plane;

    // 1) QKV projections: 3 * 32 * 512 wave-tiles / 8 waves per block
    mhca_qkv_proj<<<(3 * 32 * 512) / 8, 256, 0, stream>>>(
        q, Wq, bq, Wk, bk, Wv, bv, QH, KH, VH);

    // 2) Windowed attention: 64 * 4096 waves / 8 per block
    mhca_win_attn<<<(64 * 4096) / 8, 256, 0, stream>>>(QH, KH, VH, qmask);

    // 3) Output projection: 32 * 512 wave-tiles / 8 per block
    mhca_out_proj<<<(32 * 512) / 8, 256, 0, stream>>>(QH, Wp, bp, qmask, out);

    // 4) Tuple's second output (q_mask), if the harness allotted room for it.
    const size_t main_elems = (size_t)BS_DIM * C_DIM * T_DIM;
    if ((size_t)out_size >= main_elems + (size_t)BS_DIM * T_DIM) {
        const int n = BS_DIM * T_DIM;
        mhca_mask_copy<<<(n + 255) / 256, 256, 0, stream>>>(
            qmask, out + main_elems, n);
    }
}